// GAT_7919919693880
// MI455X (gfx1250) — compile-verified
//
#include <hip/hip_runtime.h>

#define NN       50000
#define NE       800000
#define IN_F     512
#define H_F      32
#define HEADS    8
#define HID      (H_F * HEADS)   // 256
#define N_CLS    40
#define NEG_SLOPE 0.2f

typedef __attribute__((ext_vector_type(2))) float v2f;
typedef __attribute__((ext_vector_type(8))) float v8f;

// ---------------------------------------------------------------------------
// Utility: zero-fill (bit pattern 0 == 0.0f, works for uint and float buffers)
// ---------------------------------------------------------------------------
__global__ void zero_u32(unsigned* __restrict__ p, int n) {
  int i = blockIdx.x * blockDim.x + threadIdx.x;
  if (i < n) p[i] = 0u;
}

// ---------------------------------------------------------------------------
// fp32 GEMM via V_WMMA_F32_16X16X4_F32, 2x2 register tiling.
// C[M,NC] = A[M,K] * B[K,NC], row-major, fp32 exact.
// Block = 128 threads = 4 waves; each wave computes a 32x32 C block
// (2 A fragments x 2 B fragments -> 4 accumulator tiles), so each loaded
// fragment feeds two WMMAs. Inner loop: 2x global_load_b64 (A) +
// 4x global_load_b32 (B) per 4 WMMAs, branch-free.
// Out-of-range rows/cols are clamped for loads (always valid memory) and
// guarded at the store, so EXEC stays all-ones around the WMMAs.
// ---------------------------------------------------------------------------
__global__ __launch_bounds__(128)
void gemm_wmma_f32(const float* __restrict__ A, const float* __restrict__ B,
                   float* __restrict__ C, int M, int K, int NC) {
  const int lane = threadIdx.x & 31;
  const int wv   = threadIdx.x >> 5;
  const int mt   = blockIdx.x * 4 + wv;          // 32-row tile index
  const int m0   = mt * 32;
  if (m0 >= M) return;                            // wave-uniform exit
  const int n0 = blockIdx.y * 32;
  const int mi = lane & 15;
  const int gp = lane >> 4;                       // half-wave: K pair select

  // A rows for the two fragments (clamped: garbage rows never stored)
  const int r0 = min(m0 + mi,      M - 1);
  const int r1 = min(m0 + 16 + mi, M - 1);
  // B columns for the two fragments
  const int nA = n0 + mi, nB = nA + 16;
  const int cA = min(nA, NC - 1);
  const int cB = min(nB, NC - 1);

  v8f acc00 = {}, acc01 = {}, acc10 = {}, acc11 = {};

  const float* a0p = A + (size_t)r0 * K + gp * 2;
  const float* a1p = A + (size_t)r1 * K + gp * 2;
  const float* bp  = B + (size_t)(gp * 2) * NC;

#pragma unroll 2
  for (int k = 0; k < K; k += 4) {
    v2f a0, a1, b0, b1;
    a0.x = a0p[0];     a0.y = a0p[1];
    a1.x = a1p[0];     a1.y = a1p[1];
    b0.x = bp[cA];     b0.y = bp[NC + cA];
    b1.x = bp[cB];     b1.y = bp[NC + cB];

    acc00 = __builtin_amdgcn_wmma_f32_16x16x4_f32(false, a0, false, b0, (short)0, acc00, false, false);
    acc01 = __builtin_amdgcn_wmma_f32_16x16x4_f32(false, a0, false, b1, (short)0, acc01, false, false);
    acc10 = __builtin_amdgcn_wmma_f32_16x16x4_f32(false, a1, false, b0, (short)0, acc10, false, false);
    acc11 = __builtin_amdgcn_wmma_f32_16x16x4_f32(false, a1, false, b1, (short)0, acc11, false, false);

    a0p += 4;
    a1p += 4;
    bp  += (size_t)4 * NC;
  }

  // C tile layout: lanes 0-15 rows 0-7 (vgpr v = row), lanes 16-31 rows 8-15
#pragma unroll
  for (int v = 0; v < 8; ++v) {
    const int ra = m0 + gp * 8 + v;       // rows 0..15 of the 32-row tile
    const int rb = ra + 16;               // rows 16..31
    if (ra < M) {
      float* crow = &C[(size_t)ra * NC];
      if (nA < NC) crow[nA] = acc00[v];
      if (nB < NC) crow[nB] = acc01[v];
    }
    if (rb < M) {
      float* crow = &C[(size_t)rb * NC];
      if (nA < NC) crow[nA] = acc10[v];
      if (nB < NC) crow[nB] = acc11[v];
    }
  }
}

// ---------------------------------------------------------------------------
// el[n,h] = sum_d feat[n,h,d]*al[h,d];  er likewise. One thread per (n,h).
// ---------------------------------------------------------------------------
__global__ void attn_scores(const float* __restrict__ feat,
                            const float* __restrict__ al,
                            const float* __restrict__ ar,
                            float* __restrict__ el, float* __restrict__ er,
                            int n, int H, int D) {
  int i = blockIdx.x * blockDim.x + threadIdx.x;
  if (i >= n * H) return;
  int node = i / H, h = i % H;
  const float* f = feat + (size_t)node * H * D + (size_t)h * D;
  float sl = 0.f, sr = 0.f;
  for (int d = 0; d < D; ++d) {
    float v = f[d];
    sl += v * al[h * D + d];
    sr += v * ar[h * D + d];
  }
  el[i] = sl;
  er[i] = sr;
}

// monotonic float <-> uint map so unsigned atomicMax orders like float compare
__device__ __forceinline__ unsigned fflip(float f) {
  unsigned u = __float_as_uint(f);
  return (u & 0x80000000u) ? ~u : (u | 0x80000000u);
}

// Pass 1: e = leaky_relu(el[src]+er[dst]); segment-max into mx (flipped uint)
__global__ void edge_e_max(const int* __restrict__ src, const int* __restrict__ dst,
                           const float* __restrict__ el, const float* __restrict__ er,
                           float* __restrict__ ebuf, unsigned* __restrict__ mx,
                           int E, int H) {
  int i = blockIdx.x * blockDim.x + threadIdx.x;
  if (i >= E * H) return;
  int e = i / H, h = i % H;
  float v = el[src[e] * H + h] + er[dst[e] * H + h];
  v = (v > 0.f) ? v : NEG_SLOPE * v;
  ebuf[i] = v;
  atomicMax(&mx[dst[e] * H + h], fflip(v));
}

// unflip in place; non-finite (no incoming edges) -> 0 per reference
__global__ void finalize_max(unsigned* __restrict__ mx, int n) {
  int i = blockIdx.x * blockDim.x + threadIdx.x;
  if (i >= n) return;
  unsigned u = mx[i];
  float f = (u & 0x80000000u) ? __uint_as_float(u ^ 0x80000000u)
                              : __uint_as_float(~u);
  ((float*)mx)[i] = isfinite(f) ? f : 0.f;
}

// Pass 2: ee = exp(e - m[dst]); segment-sum into den
__global__ void edge_softmax_num(const int* __restrict__ dst,
                                 const float* __restrict__ mbuf,
                                 float* __restrict__ ebuf,
                                 float* __restrict__ den, int E, int H) {
  int i = blockIdx.x * blockDim.x + threadIdx.x;
  if (i >= E * H) return;
  int e = i / H, h = i % H;
  float ee = __expf(ebuf[i] - mbuf[dst[e] * H + h]);
  ebuf[i] = ee;
  atomicAdd(&den[dst[e] * H + h], ee);
}

// Pass 3: agg[dst] += alpha * feat[src]; one thread per (edge, 4 channels).
// float4 gathers (global_load_b128) + 4x global_atomic_add_f32.
// feat fits in 192MB L2, so gathers/atomics stay on-chip.
__global__ void edge_aggregate4(const int* __restrict__ src, const int* __restrict__ dst,
                                const float* __restrict__ ebuf, const float* __restrict__ den,
                                const float* __restrict__ feat, float* __restrict__ agg,
                                int E, int H, int D) {
  const int HDq = (H * D) >> 2;                       // channel quads per edge
  long long i = (long long)blockIdx.x * blockDim.x + threadIdx.x;
  if (i >= (long long)E * HDq) return;
  int e = (int)(i / HDq);
  int c = ((int)(i % HDq)) << 2;                      // first channel of quad
  int h = c / D;                                      // D % 4 == 0: quad stays in one head
  int s = src[e], t = dst[e];
  float alpha = ebuf[(size_t)e * H + h] / fmaxf(den[t * H + h], 1e-9f);
  const float4 f = *(const float4*)&feat[(size_t)s * H * D + c];
  float* ap = &agg[(size_t)t * H * D + c];
  atomicAdd(ap + 0, alpha * f.x);
  atomicAdd(ap + 1, alpha * f.y);
  atomicAdd(ap + 2, alpha * f.z);
  atomicAdd(ap + 3, alpha * f.w);
}

// out = agg + b (+ optional ReLU), float4 vectorized
__global__ void bias_act4(const float* __restrict__ agg, const float* __restrict__ b,
                          float* __restrict__ out, int n, int HD, int relu) {
  int i = blockIdx.x * blockDim.x + threadIdx.x;
  int q = (n * HD) >> 2;
  if (i >= q) return;
  int c = (i << 2) % HD;
  float4 v = *(const float4*)&agg[(size_t)i << 2];
  v.x += b[c + 0];
  v.y += b[c + 1];
  v.z += b[c + 2];
  v.w += b[c + 3];
  if (relu) {
    v.x = fmaxf(v.x, 0.f); v.y = fmaxf(v.y, 0.f);
    v.z = fmaxf(v.z, 0.f); v.w = fmaxf(v.w, 0.f);
  }
  *(float4*)&out[(size_t)i << 2] = v;
}

// ---------------------------------------------------------------------------
// Host orchestration
// ---------------------------------------------------------------------------
static inline int ceil_div(long long a, long long b) { return (int)((a + b - 1) / b); }

extern "C" void kernel_launch(void* const* d_in, const int* in_sizes, int n_in,
                              void* d_out, int out_size, void* d_ws, size_t ws_size,
                              hipStream_t stream) {
  const float* in_feat = (const float*)d_in[0];
  const int*   src     = (const int*)d_in[1];
  const int*   dst     = (const int*)d_in[2];
  const float* W0  = (const float*)d_in[3];
  const float* al0 = (const float*)d_in[4];
  const float* ar0 = (const float*)d_in[5];
  const float* b0  = (const float*)d_in[6];
  const float* W1  = (const float*)d_in[7];
  const float* al1 = (const float*)d_in[8];
  const float* ar1 = (const float*)d_in[9];
  const float* b1  = (const float*)d_in[10];
  const float* W2  = (const float*)d_in[11];
  const float* al2 = (const float*)d_in[12];
  const float* ar2 = (const float*)d_in[13];
  const float* b2  = (const float*)d_in[14];
  float* out = (float*)d_out;

  // workspace carve-up (256B aligned)
  char* w = (char*)d_ws;
  auto carve = [&](size_t bytes) {
    void* p = (void*)w;
    w += (bytes + 255) & ~(size_t)255;
    return p;
  };
  float*    feat = (float*)carve((size_t)NN * HID * 4);   // post-GEMM features
  float*    agg  = (float*)carve((size_t)NN * HID * 4);   // aggregation target
  float*    hbuf = (float*)carve((size_t)NN * HID * 4);   // post-activation
  float*    el   = (float*)carve((size_t)NN * HEADS * 4);
  float*    er   = (float*)carve((size_t)NN * HEADS * 4);
  unsigned* mx   = (unsigned*)carve((size_t)NN * HEADS * 4);
  float*    den  = (float*)carve((size_t)NN * HEADS * 4);
  float*    ebuf = (float*)carve((size_t)NE * HEADS * 4);
  (void)ws_size; (void)n_in; (void)in_sizes; (void)out_size;

  const int TB = 256;

  auto run_layer = [&](const float* hin, int Fin, const float* W,
                       const float* al, const float* ar, const float* b,
                       int H, int D, float* layer_out, int relu) {
    const int HD = H * D;
    // GEMM: feat = hin @ W   [NN x Fin] x [Fin x HD]
    {
      int mtiles = ceil_div(NN, 32);                 // 32-row tiles per wave
      dim3 grid(ceil_div(mtiles, 4), ceil_div(HD, 32));
      gemm_wmma_f32<<<grid, 128, 0, stream>>>(hin, W, feat, NN, Fin, HD);
    }
    // attention scores
    attn_scores<<<ceil_div((long long)NN * H, TB), TB, 0, stream>>>(
        feat, al, ar, el, er, NN, H, D);
    // zero segment state + aggregation target
    zero_u32<<<ceil_div((long long)NN * H, TB), TB, 0, stream>>>(mx, NN * H);
    zero_u32<<<ceil_div((long long)NN * H, TB), TB, 0, stream>>>((unsigned*)den, NN * H);
    zero_u32<<<ceil_div((long long)NN * HD, TB), TB, 0, stream>>>((unsigned*)agg, NN * HD);
    // edge softmax (3 passes) + aggregation
    edge_e_max<<<ceil_div((long long)NE * H, TB), TB, 0, stream>>>(
        src, dst, el, er, ebuf, mx, NE, H);
    finalize_max<<<ceil_div((long long)NN * H, TB), TB, 0, stream>>>(mx, NN * H);
    edge_softmax_num<<<ceil_div((long long)NE * H, TB), TB, 0, stream>>>(
        dst, (const float*)mx, ebuf, den, NE, H);
    edge_aggregate4<<<ceil_div((long long)NE * (HD / 4), TB), TB, 0, stream>>>(
        src, dst, ebuf, den, feat, agg, NE, H, D);
    // bias + activation
    bias_act4<<<ceil_div((long long)NN * HD / 4, TB), TB, 0, stream>>>(
        agg, b, layer_out, NN, HD, relu);
  };

  run_layer(in_feat, IN_F, W0, al0, ar0, b0, HEADS, H_F, hbuf, 1);   // layer 0
  run_layer(hbuf,    HID,  W1, al1, ar1, b1, HEADS, H_F, hbuf, 1);   // layer 1
  run_layer(hbuf,    HID,  W2, al2, ar2, b2, 1,     N_CLS, out, 0);  // output
}